// SwemCat_11081015624207
// MI455X (gfx1250) — compile-verified
//
#include <hip/hip_runtime.h>
#include <cfloat>

// SWEM embedding gather + masked max/mean pooling for MI455X (gfx1250).
// Memory-bound gather-reduce: no matrix ops, so the win comes from
// wave32-coalesced b128 loads, async index staging to LDS (ASYNCcnt path),
// prefetch, and NT output stores that keep the 200MB table hot in 192MB L2.

#ifndef __has_builtin
#define __has_builtin(x) 0
#endif

#if __has_builtin(__builtin_amdgcn_global_load_async_to_lds_b32) && \
    __has_builtin(__builtin_amdgcn_s_wait_asynccnt)
#define SWEM_ASYNC_LDS 1
#else
#define SWEM_ASYNC_LDS 0
#endif

namespace {
constexpr int kTLen    = 50;
constexpr int kDLen    = 200;
constexpr int kDim     = 512;
constexpr int kThreads = 128;              // 4 waves of 32; thread t owns dims [4t,4t+4)
constexpr int kNIdx    = kTLen + kDLen;    // 250 token ids per batch row

typedef float vfloat4 __attribute__((ext_vector_type(4)));
typedef __attribute__((address_space(1))) int gas_int;   // global
typedef __attribute__((address_space(3))) int las_int;   // LDS

__device__ inline vfloat4 vmax4(vfloat4 a, vfloat4 b) {
  vfloat4 r;
  r.x = fmaxf(a.x, b.x);
  r.y = fmaxf(a.y, b.y);
  r.z = fmaxf(a.z, b.z);
  r.w = fmaxf(a.w, b.w);
  return r;
}
}  // namespace

__global__ __launch_bounds__(kThreads) void swem_pool_kernel(
    const float* __restrict__ emb,     // [VOCAB, 512] f32, row 0 is zeros
    const int*   __restrict__ title,   // [B, 50]
    const int*   __restrict__ desc,    // [B, 200]
    const int*   __restrict__ t_len,   // [B]
    const int*   __restrict__ d_len,   // [B]
    float*       __restrict__ out)     // [B, 2048] = [t_mx | d_mx | t_avg | d_avg]
{
  __shared__ int s_idx[kNIdx];

  const int b   = blockIdx.x;
  const int tid = threadIdx.x;

  // ---- Stage this batch row's 250 token ids into LDS (CDNA5 async path) ----
#if SWEM_ASYNC_LDS
  for (int base = 0; base < kNIdx; base += kThreads) {
    const int i = base + tid;
    if (i < kNIdx) {
      const int* g = (i < kTLen) ? (title + (size_t)b * kTLen + i)
                                 : (desc  + (size_t)b * kDLen + (i - kTLen));
      __builtin_amdgcn_global_load_async_to_lds_b32(
          (gas_int*)g,
          (las_int*)(&s_idx[i]),
          /*imm offset*/ 0, /*cpol*/ 0);
    }
  }
  __builtin_amdgcn_s_wait_asynccnt(0);
#else
  for (int i = tid; i < kNIdx; i += kThreads) {
    s_idx[i] = (i < kTLen) ? title[(size_t)b * kTLen + i]
                           : desc[(size_t)b * kDLen + (i - kTLen)];
  }
#endif
  __syncthreads();

  const vfloat4* __restrict__ embv = (const vfloat4*)emb;  // row r -> embv[r*128 + tid]
  const int tl = t_len[b];
  const int dl = d_len[b];

  vfloat4 tmx = {-FLT_MAX, -FLT_MAX, -FLT_MAX, -FLT_MAX};
  vfloat4 dmx = tmx;
  vfloat4 tsm = {0.f, 0.f, 0.f, 0.f};
  vfloat4 dsm = tsm;

  // ---- Title pooling: masked by looping t < tl only ----
  #pragma unroll 2
  for (int t = 0; t < tl; ++t) {
    const size_t r = (size_t)s_idx[t];
    const vfloat4 v = embv[r * (size_t)(kDim / 4) + tid];
    if (t + 2 < tl) {
      __builtin_prefetch(&embv[(size_t)s_idx[t + 2] * (size_t)(kDim / 4) + tid], 0, 3);
    }
    tmx = vmax4(tmx, v);
    tsm += v;
  }

  // ---- Description pooling ----
  #pragma unroll 2
  for (int t = 0; t < dl; ++t) {
    const size_t r = (size_t)s_idx[kTLen + t];
    const vfloat4 v = embv[r * (size_t)(kDim / 4) + tid];
    if (t + 2 < dl) {
      __builtin_prefetch(&embv[(size_t)s_idx[kTLen + t + 2] * (size_t)(kDim / 4) + tid], 0, 3);
    }
    dmx = vmax4(dmx, v);
    dsm += v;
  }

  // ---- Epilogue: [t_mx | d_mx | t_avg | d_avg], NT stores (write-once data) ----
  const vfloat4 zero = {0.f, 0.f, 0.f, 0.f};
  const vfloat4 o0 = (tl > 0) ? tmx : zero;
  const vfloat4 o1 = (dl > 0) ? dmx : zero;
  const float tinv = 1.0f / (float)((tl > 0) ? tl : 1);
  const float dinv = 1.0f / (float)((dl > 0) ? dl : 1);
  const vfloat4 o2 = tsm * tinv;
  const vfloat4 o3 = dsm * dinv;

  float* ob = out + (size_t)b * (size_t)(4 * kDim);
  const int d = tid * 4;
  __builtin_nontemporal_store(o0, (vfloat4*)(ob + d));
  __builtin_nontemporal_store(o1, (vfloat4*)(ob + kDim + d));
  __builtin_nontemporal_store(o2, (vfloat4*)(ob + 2 * kDim + d));
  __builtin_nontemporal_store(o3, (vfloat4*)(ob + 3 * kDim + d));
}

extern "C" void kernel_launch(void* const* d_in, const int* in_sizes, int n_in,
                              void* d_out, int out_size, void* d_ws, size_t ws_size,
                              hipStream_t stream) {
  (void)n_in; (void)out_size; (void)d_ws; (void)ws_size;

  const float* emb   = (const float*)d_in[0];
  const int*   title = (const int*)d_in[1];
  const int*   desc  = (const int*)d_in[2];
  const int*   t_len = (const int*)d_in[3];
  const int*   d_len = (const int*)d_in[4];
  // d_in[5] = mode (unused by the reference computation)
  float* out = (float*)d_out;

  const int B = in_sizes[3];  // element count of t_len == batch size

  swem_pool_kernel<<<dim3(B), dim3(kThreads), 0, stream>>>(
      emb, title, desc, t_len, d_len, out);
}